// TFN_72258529788075
// MI455X (gfx1250) — compile-verified
//
#include <hip/hip_runtime.h>
#include <cmath>

#define TFN_N 512
#define TFN_C 32
#define TFN_R 32
#define TFN_EPS 1e-8f

typedef float v2f __attribute__((ext_vector_type(2)));
typedef float v8f __attribute__((ext_vector_type(8)));

// Wave-local LDS ordering fence: hardware-wait all outstanding DS ops and stop
// compiler reordering across the LDS C-layout -> A-layout round trip.
__device__ __forceinline__ void lds_wave_fence() {
  asm volatile("s_wait_dscnt 0x0" ::: "memory");
}

__global__ __launch_bounds__(512, 1)
void tfn_fused_kernel(
    const float* __restrict__ in0, const float* __restrict__ in1,
    const float* __restrict__ in2, const float* __restrict__ rbf,
    const float* __restrict__ rij,
    const float* __restrict__ w1_0, const float* __restrict__ b1_0,
    const float* __restrict__ w2_0, const float* __restrict__ b2_0,
    const float* __restrict__ w1_1, const float* __restrict__ b1_1,
    const float* __restrict__ w2_1, const float* __restrict__ b2_1,
    const float* __restrict__ w1_2, const float* __restrict__ b1_2,
    const float* __restrict__ w2_2, const float* __restrict__ b2_2,
    const float* __restrict__ w1_3, const float* __restrict__ b1_3,
    const float* __restrict__ w2_3, const float* __restrict__ b2_3,
    const float* __restrict__ w1_4, const float* __restrict__ b1_4,
    const float* __restrict__ w2_4, const float* __restrict__ b2_4,
    float* __restrict__ out)
{
  __shared__ __align__(16) float s_w1[5][TFN_R * TFN_R];  // w1[j][r], row-major
  __shared__ __align__(16) float s_w2[5][TFN_C * TFN_R];  // w2[c][j], row-major
  __shared__ __align__(16) float s_b1[5][TFN_R];
  __shared__ __align__(16) float s_b2[5][TFN_C];
  // H scratch (per-wave) and final reduction buffer overlap; separated by barriers.
  __shared__ __align__(16) union {
    float H[8][16 * TFN_R];   // [wave][row(b) * 32 + col(j)]
    float red[8][16][42];     // [wave][lane%16][2 c-halves * 21 comps]
  } s_u;

  const int tid  = threadIdx.x;
  const int a    = blockIdx.x;
  const int wave = tid >> 5;
  const int lane = tid & 31;
  const int half = lane >> 4;     // WMMA lane-half (selects M+8 / K+2)
  const int l16  = lane & 15;
  const int c0   = l16;           // channel for c-tile 0
  const int c1   = 16 + l16;      // channel for c-tile 1
  const int koff = 2 * half;      // K sub-offset within a 16x4 chunk

  // ---- stage the 5 radial-MLP weight sets into LDS ----
  const float* w1p[5] = {w1_0, w1_1, w1_2, w1_3, w1_4};
  const float* b1p[5] = {b1_0, b1_1, b1_2, b1_3, b1_4};
  const float* w2p[5] = {w2_0, w2_1, w2_2, w2_3, w2_4};
  const float* b2p[5] = {b2_0, b2_1, b2_2, b2_3, b2_4};
#pragma unroll
  for (int f = 0; f < 5; ++f) {
    for (int i = tid; i < TFN_R * TFN_R; i += 512) s_w1[f][i] = w1p[f][i];
    for (int i = tid; i < TFN_C * TFN_R; i += 512) s_w2[f][i] = w2p[f][i];
    if (tid < TFN_R) s_b1[f][tid] = b1p[f][tid];
    else if (tid < TFN_R + TFN_C) s_b2[f][tid - TFN_R] = b2p[f][tid - TFN_R];
  }
  __syncthreads();

  // 21 components x 2 channel halves
  float acc[42];
#pragma unroll
  for (int i = 0; i < 42; ++i) acc[i] = 0.0f;

  float* Hbase = &s_u.H[wave][0];

  for (int bt = 0; bt < 4; ++bt) {
    const int b0 = (wave + 8 * bt) * 16;

    // A-chunks of rbf[a, b0:b0+16, 0:32] (fp32 16x4 layout: 2 floats/lane/chunk)
    const float* rbf_row = rbf + ((size_t)a * TFN_N + (size_t)(b0 + l16)) * TFN_R;
    v2f arb[8];
#pragma unroll
    for (int kc = 0; kc < 8; ++kc)
      arb[kc] = *(const v2f*)(rbf_row + 4 * kc + koff);

    if (bt < 3)  // pull next tile toward the caches (global_prefetch_b8)
      __builtin_prefetch(rbf + ((size_t)a * TFN_N + (size_t)(b0 + 128 + l16)) * TFN_R, 0, 1);

#pragma unroll
    for (int f = 0; f < 5; ++f) {
      // ---- GEMM1: H = relu(rbf @ w1^T + b1), two 16x16 j-tiles, K=32 ----
      v8f h0 = {}; v8f h1 = {};
#pragma unroll
      for (int kc = 0; kc < 8; ++kc) {
        v2f bm0 = *(const v2f*)(&s_w1[f][(size_t)(0 * 16 + l16) * TFN_R + 4 * kc + koff]);
        v2f bm1 = *(const v2f*)(&s_w1[f][(size_t)(1 * 16 + l16) * TFN_R + 4 * kc + koff]);
        h0 = __builtin_amdgcn_wmma_f32_16x16x4_f32(false, arb[kc], false, bm0,
                                                   (short)0, h0, false, false);
        h1 = __builtin_amdgcn_wmma_f32_16x16x4_f32(false, arb[kc], false, bm1,
                                                   (short)0, h1, false, false);
      }
      const float bj0 = s_b1[f][c0];
      const float bj1 = s_b1[f][c1];
      // bias + relu in C-layout, spill to LDS row-major [b-row][j]
#pragma unroll
      for (int m = 0; m < 8; ++m) {
        Hbase[(size_t)(m + 8 * half) * TFN_R + c0] = fmaxf(h0[m] + bj0, 0.0f);
        Hbase[(size_t)(m + 8 * half) * TFN_R + c1] = fmaxf(h1[m] + bj1, 0.0f);
      }
      lds_wave_fence();
      // reload H in A-layout
      v2f ah[8];
#pragma unroll
      for (int kc = 0; kc < 8; ++kc)
        ah[kc] = *(const v2f*)(Hbase + (size_t)l16 * TFN_R + 4 * kc + koff);
      lds_wave_fence();

      // ---- GEMM2: rad = H @ w2^T + b2, two 16x16 c-tiles ----
      v8f r0 = {}; v8f r1 = {};
#pragma unroll
      for (int kc = 0; kc < 8; ++kc) {
        v2f cm0 = *(const v2f*)(&s_w2[f][(size_t)c0 * TFN_R + 4 * kc + koff]);
        v2f cm1 = *(const v2f*)(&s_w2[f][(size_t)c1 * TFN_R + 4 * kc + koff]);
        r0 = __builtin_amdgcn_wmma_f32_16x16x4_f32(false, ah[kc], false, cm0,
                                                   (short)0, r0, false, false);
        r1 = __builtin_amdgcn_wmma_f32_16x16x4_f32(false, ah[kc], false, cm1,
                                                   (short)0, r1, false, false);
      }
      const float bias20 = s_b2[f][c0];
      const float bias21 = s_b2[f][c1];

      // ---- aggregate over this lane-half's 8 neighbor rows ----
#pragma unroll
      for (int m = 0; m < 8; ++m) {
        const int b = b0 + m + 8 * half;
        const float* rp = rij + ((size_t)a * TFN_N + (size_t)b) * 3;
        const float x = rp[0], y = rp[1], z = rp[2];
        const float n2  = x * x + y * y + z * z;
        const float nrm = sqrtf(n2);
        const float rv0 = r0[m] + bias20;
        const float rv1 = r1[m] + bias21;

        if (f == 0) {
          // identity CG: o0_s, o1_s, o2_s (no mask)
          const float* i0p = in0 + (size_t)b * TFN_C;
          const float* i1p = in1 + (size_t)b * TFN_C * 3;
          const float* i2p = in2 + (size_t)b * TFN_C * 5;
          acc[0]      += rv0 * i0p[c0];
          acc[21 + 0] += rv1 * i0p[c1];
#pragma unroll
          for (int i = 0; i < 3; ++i) {
            acc[5 + i]      += rv0 * i1p[c0 * 3 + i];
            acc[21 + 5 + i] += rv1 * i1p[c1 * 3 + i];
          }
#pragma unroll
          for (int i = 0; i < 5; ++i) {
            acc[16 + i]      += rv0 * i2p[c0 * 5 + i];
            acc[21 + 16 + i] += rv1 * i2p[c1 * 5 + i];
          }
        } else {
          const float maskf = (nrm >= TFN_EPS) ? 1.0f : 0.0f;
          const float rm0 = rv0 * maskf;
          const float rm1 = rv1 * maskf;
          if (f == 1) {                 // 0x1->1 : u * r01 * in0
            const float inv = 1.0f / (nrm + TFN_EPS);
            const float ux = x * inv, uy = y * inv, uz = z * inv;
            const float* i0p = in0 + (size_t)b * TFN_C;
            const float s0 = rm0 * i0p[c0];
            const float s1 = rm1 * i0p[c1];
            acc[2] += ux * s0; acc[3] += uy * s0; acc[4] += uz * s0;
            acc[21 + 2] += ux * s1; acc[21 + 3] += uy * s1; acc[21 + 4] += uz * s1;
          } else if (f == 2) {          // 0x2->2 : y2 * r02 * in0
            const float ir2 = 1.0f / fmaxf(n2, TFN_EPS);
            float y2v[5];
            y2v[0] = x * y * ir2;
            y2v[1] = y * z * ir2;
            y2v[2] = (-x * x - y * y + 2.0f * z * z) * ir2 * 0.28867513459481287f; // 1/(2*sqrt3)
            y2v[3] = z * x * ir2;
            y2v[4] = (x * x - y * y) * ir2 * 0.5f;
            const float* i0p = in0 + (size_t)b * TFN_C;
            const float s0 = rm0 * i0p[c0];
            const float s1 = rm1 * i0p[c1];
#pragma unroll
            for (int i = 0; i < 5; ++i) {
              acc[11 + i]      += y2v[i] * s0;
              acc[21 + 11 + i] += y2v[i] * s1;
            }
          } else if (f == 3) {          // 1x1->0 : r10 * (u . in1)
            const float inv = 1.0f / (nrm + TFN_EPS);
            const float ux = x * inv, uy = y * inv, uz = z * inv;
            const float* i1p = in1 + (size_t)b * TFN_C * 3;
            const float d0 = ux * i1p[c0 * 3 + 0] + uy * i1p[c0 * 3 + 1] + uz * i1p[c0 * 3 + 2];
            const float d1 = ux * i1p[c1 * 3 + 0] + uy * i1p[c1 * 3 + 1] + uz * i1p[c1 * 3 + 2];
            acc[1]      += rm0 * d0;
            acc[21 + 1] += rm1 * d1;
          } else {                      // 1x1->1 : r11 * (u x in1)
            const float inv = 1.0f / (nrm + TFN_EPS);
            const float ux = x * inv, uy = y * inv, uz = z * inv;
            const float* i1p = in1 + (size_t)b * TFN_C * 3;
            const float wx0 = i1p[c0 * 3 + 0], wy0 = i1p[c0 * 3 + 1], wz0 = i1p[c0 * 3 + 2];
            const float wx1 = i1p[c1 * 3 + 0], wy1 = i1p[c1 * 3 + 1], wz1 = i1p[c1 * 3 + 2];
            acc[8]  += rm0 * (uy * wz0 - uz * wy0);
            acc[9]  += rm0 * (uz * wx0 - ux * wz0);
            acc[10] += rm0 * (ux * wy0 - uy * wx0);
            acc[21 + 8]  += rm1 * (uy * wz1 - uz * wy1);
            acc[21 + 9]  += rm1 * (uz * wx1 - ux * wz1);
            acc[21 + 10] += rm1 * (ux * wy1 - uy * wx1);
          }
        }
      }
    }
  }

  // combine the two lane-halves (same channels, disjoint b sets)
#pragma unroll
  for (int i = 0; i < 42; ++i)
    acc[i] += __shfl_xor(acc[i], 16, 32);

  __syncthreads();               // all waves done with s_u.H before red overlays it
  if (half == 0) {
#pragma unroll
    for (int i = 0; i < 42; ++i)
      s_u.red[wave][l16][i] = acc[i];
  }
  __syncthreads();

  // deterministic cross-wave reduce + store: 32 channels * 21 comps = 672 outputs per a
  for (int t = tid; t < TFN_C * 21; t += 512) {
    const int c    = t / 21;
    const int comp = t % 21;
    const int ridx = comp + 21 * (c >> 4);
    const int lidx = c & 15;
    float s = 0.0f;
#pragma unroll
    for (int w = 0; w < 8; ++w) s += s_u.red[w][lidx][ridx];

    const size_t NC = (size_t)TFN_N * TFN_C;
    size_t o;
    if (comp == 0)      o = (size_t)a * TFN_C + c;                                      // o0_s
    else if (comp == 1) o = NC + (size_t)a * TFN_C + c;                                 // o0_b
    else if (comp < 5)  o = 2 * NC + ((size_t)(0 * TFN_N + a) * TFN_C + c) * 3 + (comp - 2);   // o1_a
    else if (comp < 8)  o = 2 * NC + ((size_t)(1 * TFN_N + a) * TFN_C + c) * 3 + (comp - 5);   // o1_s
    else if (comp < 11) o = 2 * NC + ((size_t)(2 * TFN_N + a) * TFN_C + c) * 3 + (comp - 8);   // o1_c
    else if (comp < 16) o = 2 * NC + 9 * NC + ((size_t)(0 * TFN_N + a) * TFN_C + c) * 5 + (comp - 11); // o2_a
    else                o = 2 * NC + 9 * NC + ((size_t)(1 * TFN_N + a) * TFN_C + c) * 5 + (comp - 16); // o2_s
    out[o] = s;
  }
}

extern "C" void kernel_launch(void* const* d_in, const int* in_sizes, int n_in,
                              void* d_out, int out_size, void* d_ws, size_t ws_size,
                              hipStream_t stream) {
  (void)in_sizes; (void)n_in; (void)out_size; (void)d_ws; (void)ws_size;
  const float* in0 = (const float*)d_in[0];
  const float* in1 = (const float*)d_in[1];
  const float* in2 = (const float*)d_in[2];
  const float* rbf = (const float*)d_in[3];
  const float* rij = (const float*)d_in[4];
  tfn_fused_kernel<<<TFN_N, 512, 0, stream>>>(
      in0, in1, in2, rbf, rij,
      (const float*)d_in[5],  (const float*)d_in[6],  (const float*)d_in[7],  (const float*)d_in[8],   // f00
      (const float*)d_in[9],  (const float*)d_in[10], (const float*)d_in[11], (const float*)d_in[12],  // f01
      (const float*)d_in[13], (const float*)d_in[14], (const float*)d_in[15], (const float*)d_in[16],  // f02
      (const float*)d_in[17], (const float*)d_in[18], (const float*)d_in[19], (const float*)d_in[20],  // f10
      (const float*)d_in[21], (const float*)d_in[22], (const float*)d_in[23], (const float*)d_in[24],  // f11
      (float*)d_out);
}